// SurvivalCIndexLoss_6047313953157
// MI455X (gfx1250) — compile-verified
//
#include <hip/hip_runtime.h>

typedef float v2f __attribute__((ext_vector_type(2)));
typedef float v8f __attribute__((ext_vector_type(8)));

#define JSTRIP 2048
#define JTILES (JSTRIP / 16)

__global__ __launch_bounds__(256)
void cindex_pairs_kernel(const float* __restrict__ risk,
                         const float* __restrict__ timev,
                         const int* __restrict__ event,
                         int n, unsigned* __restrict__ ws, int jStrips)
{
    __shared__ float t_s[JSTRIP];
    __shared__ float r_s[JSTRIP];
    __shared__ unsigned maskj_s[JTILES];   // low16: j-valid bits, high16: event bits
    __shared__ unsigned eim_s[8];          // per-wave i-tile event masks
    __shared__ unsigned red_s[768];

    const int tid   = threadIdx.x;
    const int bi    = blockIdx.x / jStrips;
    const int js    = blockIdx.x % jStrips;
    const int jbase = js * JSTRIP;

    // ---- stage j-strip (t, r) into LDS ----
    for (int idx = tid; idx < JSTRIP; idx += 256) {
        int j = jbase + idx;
        bool v = (j < n);
        t_s[idx] = v ? timev[j] : 3.0e38f;
        r_s[idx] = v ? risk[j]  : 0.0f;
    }
    // ---- per-j-tile packed event/valid masks ----
    if (tid < JTILES) {
        unsigned jv = 0, ej = 0;
        int jb = jbase + tid * 16;
        #pragma unroll
        for (int k = 0; k < 16; ++k) {
            int j = jb + k;
            if (j < n) { jv |= (1u << k); if (event[j] == 1) ej |= (1u << k); }
        }
        maskj_s[tid] = jv | (ej << 16);
    }
    // ---- per-i-tile event masks (invalid i -> bit 0 kills all contributions) ----
    if (tid < 8) {
        unsigned m = 0;
        int ib = bi * 128 + tid * 16;
        #pragma unroll
        for (int k = 0; k < 16; ++k) {
            int ig = ib + k;
            if (ig < n && event[ig] == 1) m |= (1u << k);
        }
        eim_s[tid] = m;
    }
    __syncthreads();

    const int wid  = tid >> 5;      // wave32: 8 waves per block
    const int lane = tid & 31;
    const int i0   = bi * 128 + wid * 16;

    unsigned conc = 0, tied = 0, comp = 0;

    if (i0 < n) {                   // wave-uniform
        const bool lo = (lane < 16);
        const int  nl = lane & 15;
        const int  ig = i0 + nl;
        const bool iv = lo && (ig < n);
        const float ri = iv ? risk[ig]  : 0.0f;
        const float ti = iv ? timev[ig] : 0.0f;
        // A (16x4 f32): lanes 0-15 hold K=0 (v0) and K=1 (v1); lanes 16-31 hold K=2,3 = 0
        v2f Ar; Ar.x = lo ? ri : 0.0f; Ar.y = lo ? 1.0f : 0.0f;
        v2f At; At.x = lo ? ti : 0.0f; At.y = lo ? 1.0f : 0.0f;
        const unsigned eim = eim_s[wid];
        const int hi8 = (lane >> 4) << 3;   // M offset: 0 for lanes 0-15, 8 for 16-31

        for (int jt = 0; jt < JTILES; ++jt) {   // wave-uniform loop & skips
            int j0 = jbase + jt * 16;
            if (j0 >= n) break;
            if (j0 < i0) continue;              // tile-aligned: entirely lower triangle
            const bool diag = (j0 == i0);

            // B (4x16 f32): row K=0 = ones, row K=1 = -x_j, rows 2,3 = 0
            v2f Br; Br.x = lo ? 1.0f : 0.0f; Br.y = lo ? -r_s[jt * 16 + nl] : 0.0f;
            v2f Bt; Bt.x = Br.x;             Bt.y = lo ? -t_s[jt * 16 + nl] : 0.0f;
            v8f z = {};
            // D[m][n] = x_i[m] - x_j[n]  (exact f32 subtraction semantics)
            v8f Dr = __builtin_amdgcn_wmma_f32_16x16x4_f32(
                false, Ar, false, Br, (short)0, z, false, false);
            v8f Dt = __builtin_amdgcn_wmma_f32_16x16x4_f32(
                false, At, false, Bt, (short)0, z, false, false);

            unsigned pk  = maskj_s[jt];
            unsigned jvm = pk & 0xFFFFu;
            unsigned ejm = pk >> 16;
            unsigned ej  = (ejm >> nl) & 1u;
            unsigned jv  = (jvm >> nl) & 1u;

            #pragma unroll
            for (int v = 0; v < 8; ++v) {
                int m = v + hi8;
                float dr = Dr[v];
                float dt = Dt[v];
                unsigned ei  = (eim >> m) & 1u;
                unsigned up  = (!diag || (nl > m)) ? 1u : 0u;
                unsigned mA  = ei & (ej ^ 1u) & jv & (dt < 0.0f ? 1u : 0u);
                unsigned mB  = ei & ej;
                unsigned mAB = (mA | mB) & up;
                comp += mAB;
                unsigned cB = ((dt < 0.0f && dr > 0.0f) ||
                               (dt > 0.0f && dr < 0.0f)) ? 1u : 0u;
                conc += ((mA & (dr > 0.0f ? 1u : 0u)) | (mB & cB)) & up;
                tied += mAB & (dr == 0.0f ? 1u : 0u);
            }
        }
    }

    // ---- block reduction of the three u32 counters ----
    red_s[tid] = conc; red_s[256 + tid] = tied; red_s[512 + tid] = comp;
    __syncthreads();
    for (int s = 128; s > 0; s >>= 1) {
        if (tid < s) {
            red_s[tid]       += red_s[tid + s];
            red_s[256 + tid] += red_s[256 + tid + s];
            red_s[512 + tid] += red_s[512 + tid + s];
        }
        __syncthreads();
    }
    if (tid == 0) {
        ws[blockIdx.x * 3 + 0] = red_s[0];
        ws[blockIdx.x * 3 + 1] = red_s[256];
        ws[blockIdx.x * 3 + 2] = red_s[512];
    }
}

__global__ __launch_bounds__(256)
void cindex_final_kernel(const unsigned* __restrict__ ws, int nPart,
                         float* __restrict__ out)
{
    __shared__ unsigned red_s[768];
    int tid = threadIdx.x;
    unsigned conc = 0, tied = 0, comp = 0;
    for (int b = tid; b < nPart; b += 256) {
        conc += ws[b * 3 + 0];
        tied += ws[b * 3 + 1];
        comp += ws[b * 3 + 2];
    }
    red_s[tid] = conc; red_s[256 + tid] = tied; red_s[512 + tid] = comp;
    __syncthreads();
    for (int s = 128; s > 0; s >>= 1) {
        if (tid < s) {
            red_s[tid]       += red_s[tid + s];
            red_s[256 + tid] += red_s[256 + tid + s];
            red_s[512 + tid] += red_s[512 + tid + s];
        }
        __syncthreads();
    }
    if (tid == 0) {
        float concf = (float)red_s[0];
        float tiedf = 0.5f * (float)red_s[256];
        float compf = (float)red_s[512];
        float c = (compf > 0.0f) ? ((concf + tiedf) / fmaxf(compf, 1.0f)) : 0.5f;
        out[0] = 1.0f - c;
    }
}

extern "C" void kernel_launch(void* const* d_in, const int* in_sizes, int n_in,
                              void* d_out, int out_size, void* d_ws, size_t ws_size,
                              hipStream_t stream)
{
    const float* risk  = (const float*)d_in[0];   // (B,1) f32
    const float* timev = (const float*)d_in[1];   // (B,)  f32
    const int*   ev    = (const int*)d_in[2];     // (B,)  i32
    float* out = (float*)d_out;
    unsigned* ws = (unsigned*)d_ws;

    int n = in_sizes[1];
    int iBlocks = (n + 127) / 128;
    int jStrips = (n + JSTRIP - 1) / JSTRIP;
    int nPart = iBlocks * jStrips;   // 64 * 4 = 256 for B=8192

    cindex_pairs_kernel<<<dim3(nPart), dim3(256), 0, stream>>>(
        risk, timev, ev, n, ws, jStrips);
    cindex_final_kernel<<<dim3(1), dim3(256), 0, stream>>>(ws, nPart, out);
}